// IntraTemporalAttention_18141941858855
// MI455X (gfx1250) — compile-verified
//
#include <hip/hip_runtime.h>
#include <math.h>

#define SEQ    4096
#define BATCH  16
#define HIDDEN 512

typedef __attribute__((ext_vector_type(2))) float v2f;
typedef __attribute__((ext_vector_type(8))) float v8f;

// Hardware transcendental tanh (CDNA5 TRANS op). The ISA requires one
// independent instruction (or V_NOP) after a TRANS op before its result is
// consumed; bake the V_NOP into the asm so scheduling can't break it.
__device__ __forceinline__ float htanh(float x) {
    float r;
    asm("v_tanh_f32 %0, %1\n\tv_nop" : "=v"(r) : "v"(x));
    return r;
}

// ---------------------------------------------------------------------------
// Kernel 1: decoder_features[16][512] = outputs_hidden[16][512] x W_feat^T + b
// One wave (32 threads) per 16-wide N tile; K=512 consumed 4 at a time with
// chained V_WMMA_F32_16X16X4_F32 accumulation (exact f32, matches reference).
// A (16x4 f32): lanes 0-15 hold M=lane with K={k0,k0+1}; lanes 16-31 K={k0+2,k0+3}.
// B (4x16 f32): lanes 0-15 hold N=lane with K={k0,k0+1}; lanes 16-31 K={k0+2,k0+3}.
// C/D (16x16 f32): VGPR r holds M=r (lanes 0-15) / M=8+r (lanes 16-31), N=lane&15.
// ---------------------------------------------------------------------------
__global__ void k_decfeat(const float* __restrict__ OH, const float* __restrict__ W,
                          const float* __restrict__ bfeat, float* __restrict__ dec) {
    const int tile = blockIdx.x;          // 0..31  (N tile of 16)
    const int lane = threadIdx.x;         // 0..31
    const int half = lane >> 4;           // 0 or 1
    const int l    = lane & 15;
    const int n    = tile * 16 + l;       // output feature index

    v8f c = {};
    for (int k0 = 0; k0 < HIDDEN; k0 += 4) {
        const int ka = k0 + half * 2;
        v2f a, b;
        a.x = OH[l * HIDDEN + ka];        // A[m=l][ka]
        a.y = OH[l * HIDDEN + ka + 1];
        b.x = W[n * HIDDEN + ka];         // B[ka][n] = W_feat[n][ka]
        b.y = W[n * HIDDEN + ka + 1];
        c = __builtin_amdgcn_wmma_f32_16x16x4_f32(
                /*neg_a=*/false, a, /*neg_b=*/false, b,
                /*c_mod=*/(short)0, c, /*reuse_a=*/false, /*reuse_b=*/false);
    }

    const float bias = bfeat[n];
#pragma unroll
    for (int r = 0; r < 8; ++r) {
        const int m = half * 8 + r;       // batch row
        dec[m * HIDDEN + n] = c[r] + bias;
    }
}

// ---------------------------------------------------------------------------
// Kernel 2: streaming pass over encoder_features (128 MB).
// One wave per (s,b) pair: each lane reads 16 floats (4x float4 -> b128),
// computes sum tanh(dec+enc)*v with hardware V_TANH_F32 (keeps the loop
// bandwidth-bound instead of VALU-bound), wave32 shfl-xor reduction,
// lane0 does exp / running-sum bookkeeping.
// ---------------------------------------------------------------------------
__global__ void k_scores(const float* __restrict__ encf, const float* __restrict__ dec,
                         const float* __restrict__ vattn, const float* __restrict__ tss,
                         const float* __restrict__ mask, float* __restrict__ attw,
                         float* __restrict__ newsum_out) {
    const int wave = (int)((blockIdx.x * blockDim.x + threadIdx.x) >> 5);
    const int lane = threadIdx.x & 31;
    const int s = wave >> 4;              // 0..4095
    const int b = wave & 15;              // 0..15

    const float* e = encf + ((size_t)s * BATCH + b) * HIDDEN;
    const float* d = dec + b * HIDDEN;

    float acc = 0.f;
#pragma unroll
    for (int ch = 0; ch < 4; ++ch) {
        const int h = ch * 128 + lane * 4;
        const float4 ev = *(const float4*)(e + h);
        const float4 dv = *(const float4*)(d + h);
        const float4 vv = *(const float4*)(vattn + h);
        acc += htanh(dv.x + ev.x) * vv.x;
        acc += htanh(dv.y + ev.y) * vv.y;
        acc += htanh(dv.z + ev.z) * vv.z;
        acc += htanh(dv.w + ev.w) * vv.w;
    }
#pragma unroll
    for (int off = 16; off > 0; off >>= 1)
        acc += __shfl_xor(acc, off, 32);

    if (lane == 0) {
        const float ta = expf(acc);                 // temporal_attention
        const float t  = tss[b * SEQ + s];
        const float m  = mask[b * SEQ + s];
        attw[b * SEQ + s]       = ta / t * m;       // pre-normalized attention
        newsum_out[b * SEQ + s] = ta + t;           // new_scores_sum (final output)
    }
}

// ---------------------------------------------------------------------------
// Kernel 3: per-batch normalization of attention (16 x 4096, tiny).
// ---------------------------------------------------------------------------
__global__ void k_norm(const float* __restrict__ attw, float* __restrict__ att_out) {
    __shared__ float red[256];
    const int b = blockIdx.x;
    const int t = threadIdx.x;

    float s = 0.f;
    for (int i = t; i < SEQ; i += 256) s += attw[b * SEQ + i];
    red[t] = s;
    __syncthreads();
    for (int off = 128; off > 0; off >>= 1) {
        if (t < off) red[t] += red[t + off];
        __syncthreads();
    }
    const float inv = 1.0f / red[0];
    for (int i = t; i < SEQ; i += 256)
        att_out[b * SEQ + i] = attw[b * SEQ + i] * inv;
}

// ---------------------------------------------------------------------------
// Kernel 4: streaming pass over encoder_out (128 MB): deterministic partial
// context sums. Block = (b, s-chunk of 128); 128 threads cover all 512 h via
// float4. Coalesced 2 KB row per iteration + global_prefetch_b8 lookahead.
// ---------------------------------------------------------------------------
__global__ void k_ctx_part(const float* __restrict__ enc, const float* __restrict__ att,
                           float* __restrict__ part) {
    const int b     = blockIdx.x & 15;
    const int chunk = blockIdx.x >> 4;    // 0..31
    const int t     = threadIdx.x;        // 0..127
    const int h     = t * 4;
    const int s0    = chunk * 128;

    float4 acc = make_float4(0.f, 0.f, 0.f, 0.f);
    for (int i = 0; i < 128; ++i) {
        const int s = s0 + i;
        if (i + 8 < 128)
            __builtin_prefetch(enc + ((size_t)(s + 8) * BATCH + b) * HIDDEN + h, 0, 0);
        const float  a  = att[b * SEQ + s];
        const float4 ev = *(const float4*)(enc + ((size_t)s * BATCH + b) * HIDDEN + h);
        acc.x += a * ev.x;
        acc.y += a * ev.y;
        acc.z += a * ev.z;
        acc.w += a * ev.w;
    }
    *(float4*)(part + ((size_t)chunk * BATCH + b) * HIDDEN + h) = acc;
}

// ---------------------------------------------------------------------------
// Kernel 5: reduce 32 partial chunks -> context[1][16][512] (tiny, exact order).
// ---------------------------------------------------------------------------
__global__ void k_ctx_reduce(const float* __restrict__ part, float* __restrict__ ctx) {
    const int b = blockIdx.x;    // 16
    const int t = threadIdx.x;   // 128
    const int h = t * 4;
    float4 acc = make_float4(0.f, 0.f, 0.f, 0.f);
    for (int c = 0; c < 32; ++c) {
        const float4 p = *(const float4*)(part + ((size_t)c * BATCH + b) * HIDDEN + h);
        acc.x += p.x; acc.y += p.y; acc.z += p.z; acc.w += p.w;
    }
    *(float4*)(ctx + b * HIDDEN + h) = acc;
}

// ---------------------------------------------------------------------------
extern "C" void kernel_launch(void* const* d_in, const int* in_sizes, int n_in,
                              void* d_out, int out_size, void* d_ws, size_t ws_size,
                              hipStream_t stream) {
    const float* OH    = (const float*)d_in[0];   // outputs_hidden   [1,16,512]
    const float* enc   = (const float*)d_in[1];   // encoder_out      [4096,16,512]
    const float* encf  = (const float*)d_in[2];   // encoder_features [4096,16,512]
    const float* mask  = (const float*)d_in[3];   // encoder_mask     [16,1,4096]
    const float* tss   = (const float*)d_in[4];   // temporal_scores_sum [16,1,4096]
    const float* W     = (const float*)d_in[5];   // W_feat           [512,512]
    const float* bfeat = (const float*)d_in[6];   // b_feat           [512]
    const float* vattn = (const float*)d_in[7];   // v_attn           [512]

    float* out = (float*)d_out;
    float* ctx = out;                              // context        [1,16,512]   (8192)
    float* att = out + BATCH * HIDDEN;             // attention      [16,1,4096]  (65536)
    float* nss = att + BATCH * SEQ;                // new_scores_sum [16,1,4096]  (65536)

    char*  ws   = (char*)d_ws;
    float* dec  = (float*)ws;                                  //  32 KB
    float* attw = (float*)(ws + 32768);                        // 256 KB
    float* part = (float*)(ws + 32768 + 262144);               //   1 MB

    // 1) tiny f32 WMMA GEMM: decoder features
    k_decfeat<<<HIDDEN / 16, 32, 0, stream>>>(OH, W, bfeat, dec);
    // 2) stream encoder_features: additive scores -> exp -> temporal bookkeeping
    k_scores<<<(SEQ * BATCH) / 8, 256, 0, stream>>>(encf, dec, vattn, tss, mask, attw, nss);
    // 3) normalize attention per batch
    k_norm<<<BATCH, 256, 0, stream>>>(attw, att);
    // 4) stream encoder_out: partial context sums (deterministic, no atomics)
    k_ctx_part<<<BATCH * 32, 128, 0, stream>>>(enc, att, part);
    // 5) reduce partials -> context
    k_ctx_reduce<<<BATCH, 128, 0, stream>>>(part, ctx);
}